// FractalLoss_24730421690932
// MI455X (gfx1250) — compile-verified
//
#include <hip/hip_runtime.h>
#include <math.h>

typedef __attribute__((ext_vector_type(4))) float v4f;
typedef __attribute__((ext_vector_type(2))) float v2f;
typedef __attribute__((ext_vector_type(8))) float v8f;

#define GRID_MAIN 2048
#define BLOCK_MAIN 256

// XOR lane exchange via ds_swizzle_b32 group-of-32 mode:
// offset = and_mask(0x1f)<<10 | or_mask(0)<<5 | xor_mask  => 0x7C00 | xor
__device__ __forceinline__ float swz_xor(float x, int xorMask) {
  int i = __float_as_int(x);
  int r;
  switch (xorMask) {           // ds_swizzle needs an immediate offset
    case 8:  r = __builtin_amdgcn_ds_swizzle(i, 0x7C08); break;
    case 4:  r = __builtin_amdgcn_ds_swizzle(i, 0x7C04); break;
    case 2:  r = __builtin_amdgcn_ds_swizzle(i, 0x7C02); break;
    default: r = __builtin_amdgcn_ds_swizzle(i, 0x7C01); break;
  }
  return __int_as_float(r);
}

// Exact full-wave (32-lane) sum using V_WMMA_F32_16X16X4_F32.
// A(16x4) = all ones, B(4x16) = one value per lane in VGPR0, VGPR1 = 0.
// D[m][n] = column-sum(n) = v[n] + v[n+16]  (layout-robust: independent of
// which K-rows map to which lane half). Finish 16->1 with ds_swizzle XORs.
__device__ __forceinline__ float wave_sum_wmma(float v) {
  v2f a; a[0] = 1.0f; a[1] = 1.0f;
  v2f b; b[0] = v;    b[1] = 0.0f;
  v8f c = {};
  v8f d = __builtin_amdgcn_wmma_f32_16x16x4_f32(false, a, false, b,
                                                (short)0, c, false, false);
  float s = d[0];                 // lane L holds v[L&15] + v[(L&15)+16]
  s += swz_xor(s, 8);
  s += swz_xor(s, 4);
  s += swz_xor(s, 2);
  s += swz_xor(s, 1);
  return s;                       // full wave sum, uniform across lanes
}

// ---------------- Kernel 1: streaming BCE / Dice partial sums ----------------
__global__ void __launch_bounds__(BLOCK_MAIN)
fl_main_reduce(const float* __restrict__ pred, const float* __restrict__ target,
               float* __restrict__ partials, int n4) {
  const v4f* p4 = (const v4f*)pred;
  const v4f* t4 = (const v4f*)target;
  float sb = 0.0f, si = 0.0f, su = 0.0f;
  const int stride = gridDim.x * blockDim.x;
  for (int i = blockIdx.x * blockDim.x + threadIdx.x; i < n4; i += stride) {
    __builtin_prefetch(&p4[i + stride], 0, 3);   // global_prefetch_b8
    __builtin_prefetch(&t4[i + stride], 0, 3);
    v4f p = p4[i];
    v4f t = t4[i];
#pragma unroll
    for (int j = 0; j < 4; ++j) {
      float pj = p[j], tj = t[j];
      // One transcendental exp per element:
      //   e = exp(-|p|);  sigmoid(p) = (p>=0 ? 1 : e) / (1+e)
      //   softplus(p) = logaddexp(0,p) = max(p,0) + log(1+e)
      float e   = __expf(-fabsf(pj));
      float r   = __builtin_amdgcn_rcpf(1.0f + e);   // v_rcp_f32
      float sig = (pj >= 0.0f ? 1.0f : e) * r;
      float sp  = fmaxf(pj, 0.0f) + __logf(1.0f + e);
      sb += sp - pj * tj;
      si += sig * tj;
      su += sig + tj;
    }
  }
  // All threads converge here: EXEC is all-ones for WMMA.
  float wb = wave_sum_wmma(sb);
  float wi = wave_sum_wmma(si);
  float wu = wave_sum_wmma(su);
  __shared__ float lds[BLOCK_MAIN / 32][3];
  const int lane = threadIdx.x & 31, wid = threadIdx.x >> 5;
  if (lane == 0) { lds[wid][0] = wb; lds[wid][1] = wi; lds[wid][2] = wu; }
  __syncthreads();
  if (threadIdx.x == 0) {
    float a = 0.0f, b = 0.0f, c = 0.0f;
    for (int w = 0; w < BLOCK_MAIN / 32; ++w) {
      a += lds[w][0]; b += lds[w][1]; c += lds[w][2];
    }
    partials[blockIdx.x * 3 + 0] = a;
    partials[blockIdx.x * 3 + 1] = b;
    partials[blockIdx.x * 3 + 2] = c;
  }
}

// ---------------- Kernel 2: box-counting fractal dimension ----------------
// blockIdx.x in [0,8): 0-3 -> target[b] (>0.5), 4-7 -> sigmoid(pred[b]) (>0.5
// i.e. pred > 0). One 512x512 image per block; 16x16-pixel tiles as 16-bit
// row masks; box counts at b=2,4,8,16 via bit folds + popcount.
__global__ void __launch_bounds__(256)
fl_fractal(const float* __restrict__ pred, const float* __restrict__ target,
           float* __restrict__ dims) {
  const int img = blockIdx.x;
  const bool isT = img < 4;
  const float* base = isT ? (target + (size_t)img * 512 * 512)
                          : (pred + (size_t)(img - 4) * 512 * 512);
  const float th = isT ? 0.5f : 0.0f;

  __shared__ unsigned cnt[5];   // nset, c2, c4, c8, c16
  if (threadIdx.x < 5) cnt[threadIdx.x] = 0u;
  __syncthreads();

  unsigned nset = 0, c2 = 0, c4 = 0, c8 = 0, c16 = 0;
  for (int tile = threadIdx.x; tile < 1024; tile += blockDim.x) {
    const int ty = tile >> 5, tx = tile & 31;
    const float* tp = base + (size_t)ty * 16 * 512 + tx * 16;
    unsigned rows[16];
#pragma unroll
    for (int r = 0; r < 16; ++r) {
      const v4f* rp = (const v4f*)(tp + (size_t)r * 512);
      unsigned m = 0;
#pragma unroll
      for (int q = 0; q < 4; ++q) {
        v4f x = rp[q];
#pragma unroll
        for (int j = 0; j < 4; ++j)
          if (x[j] > th) m |= 1u << (q * 4 + j);
      }
      rows[r] = m;
      nset += __popc(m);
    }
#pragma unroll
    for (int g = 0; g < 8; ++g) {           // 2x2 boxes
      unsigned o = rows[2 * g] | rows[2 * g + 1];
      c2 += __popc((o | (o >> 1)) & 0x5555u);
    }
#pragma unroll
    for (int g = 0; g < 4; ++g) {           // 4x4 boxes
      unsigned o = rows[4 * g] | rows[4 * g + 1] | rows[4 * g + 2] | rows[4 * g + 3];
      unsigned f = o | (o >> 1); f = (f | (f >> 2)) & 0x1111u;
      c4 += __popc(f);
    }
#pragma unroll
    for (int g = 0; g < 2; ++g) {           // 8x8 boxes
      unsigned o = 0;
#pragma unroll
      for (int r = 0; r < 8; ++r) o |= rows[8 * g + r];
      unsigned f = o | (o >> 1); f |= f >> 2; f = (f | (f >> 4)) & 0x0101u;
      c8 += __popc(f);
    }
    {
      unsigned o = 0;                       // 16x16 box
#pragma unroll
      for (int r = 0; r < 16; ++r) o |= rows[r];
      c16 += (o != 0u) ? 1u : 0u;
    }
  }
  atomicAdd(&cnt[0], nset);
  atomicAdd(&cnt[1], c2);
  atomicAdd(&cnt[2], c4);
  atomicAdd(&cnt[3], c8);
  atomicAdd(&cnt[4], c16);
  __syncthreads();

  if (threadIdx.x == 0) {
    const float cc[4] = {(float)cnt[1], (float)cnt[2], (float)cnt[3], (float)cnt[4]};
    const float lx[4] = {0.69314718f, 1.38629436f, 2.07944154f, 2.77258872f};
    float sx = 0.0f, sy = 0.0f, sxy = 0.0f, sxx = 0.0f;
#pragma unroll
    for (int k = 0; k < 4; ++k) {
      float ly = __logf(cc[k] + 1e-10f);
      sx += lx[k]; sy += ly; sxy += lx[k] * ly; sxx += lx[k] * lx[k];
    }
    float slope = (4.0f * sxy - sx * sy) / (4.0f * sxx - sx * sx);
    bool valid = (cnt[0] > 0u) && (cnt[1] > 0u) && (cnt[2] > 0u) &&
                 (cnt[3] > 0u) && (cnt[4] > 0u);
    dims[img] = valid ? -slope : 0.0f;
  }
}

// ---------------- Kernel 3: final combine ----------------
__global__ void __launch_bounds__(256)
fl_finalize(const float* __restrict__ partials, const float* __restrict__ dims,
            float* __restrict__ out, int n) {
  float sb = 0.0f, si = 0.0f, su = 0.0f;
  for (int k = threadIdx.x; k < GRID_MAIN; k += 256) {
    sb += partials[k * 3 + 0];
    si += partials[k * 3 + 1];
    su += partials[k * 3 + 2];
  }
  float wb = wave_sum_wmma(sb);
  float wi = wave_sum_wmma(si);
  float wu = wave_sum_wmma(su);
  __shared__ float lds[8][3];
  const int lane = threadIdx.x & 31, wid = threadIdx.x >> 5;
  if (lane == 0) { lds[wid][0] = wb; lds[wid][1] = wi; lds[wid][2] = wu; }
  __syncthreads();
  if (threadIdx.x == 0) {
    float a = 0.0f, b = 0.0f, c = 0.0f;
    for (int w = 0; w < 8; ++w) { a += lds[w][0]; b += lds[w][1]; c += lds[w][2]; }
    float N = (float)n;
    float bce  = a / N;
    float dice = 1.0f - 2.0f * b / (c + 1e-8f);
    float fr   = 0.25f * (fabsf(dims[0] - dims[4]) + fabsf(dims[1] - dims[5]) +
                          fabsf(dims[2] - dims[6]) + fabsf(dims[3] - dims[7]));
    out[0] = 0.3f * bce + 0.2f * dice + 0.5f * fr;
  }
}

extern "C" void kernel_launch(void* const* d_in, const int* in_sizes, int n_in,
                              void* d_out, int out_size, void* d_ws, size_t ws_size,
                              hipStream_t stream) {
  const float* pred   = (const float*)d_in[0];
  const float* target = (const float*)d_in[1];
  float* out = (float*)d_out;
  float* ws  = (float*)d_ws;
  float* partials = ws;                   // 3 * GRID_MAIN floats
  float* dims     = ws + 3 * GRID_MAIN;   // 8 floats
  const int n  = in_sizes[0];             // 64*1*512*512
  const int n4 = n / 4;

  fl_main_reduce<<<GRID_MAIN, BLOCK_MAIN, 0, stream>>>(pred, target, partials, n4);
  fl_fractal<<<8, 256, 0, stream>>>(pred, target, dims);
  fl_finalize<<<1, 256, 0, stream>>>(partials, dims, out, n);
}